// GraphSAGE_75496935129722
// MI455X (gfx1250) — compile-verified
//
#include <hip/hip_runtime.h>
#include <hip/hip_bf16.h>

// ---------------------------------------------------------------------------
// GraphSAGE (2x SAGEConv mean-agg + global mean pool) for MI455X / gfx1250.
// fp32 end-to-end; GEMMs on V_WMMA_F32_16X16X4_F32 (wave32 matrix pipe).
// ---------------------------------------------------------------------------

#define N_NODES  100000
#define N_EDGES  1000000
#define IN_CH    64
#define HID_CH   64
#define OUT_CH   128
#define N_GRAPHS 128

typedef float v2f __attribute__((ext_vector_type(2)));
typedef float v8f __attribute__((ext_vector_type(8)));

#if defined(__has_builtin)
#  if __has_builtin(__builtin_amdgcn_wmma_f32_16x16x4_f32)
#    define HAVE_WMMA_F32X4 1
#  endif
#endif

__device__ __forceinline__ v8f wmma_f32_16x16x4(v2f a, v2f b, v8f c) {
#ifdef HAVE_WMMA_F32X4
  // 8 args: (neg_a, A, neg_b, B, c_mod, C, reuse_a, reuse_b)
  return __builtin_amdgcn_wmma_f32_16x16x4_f32(
      false, a, false, b, (short)0, c, false, false);
#else
  // compile-only fallback (host pass); numerics not used on device
  c[0] = __builtin_fmaf(a.x, b.x, __builtin_fmaf(a.y, b.y, c[0]));
  return c;
#endif
}

// LDS weight layout: (k,k+1) value pairs stored contiguously per column so a
// B fragment is one aligned 8-byte ds load directly into the WMMA source
// pair. Pair-row stride chosen ≡16 (mod 32) pairs: lanes 0-15 (K-half 0) hit
// banks 0-31, lanes 16-31 (K-half 1) hit banks 32-63 -> conflict-free.
#define PSTRIDE1 80    // 64 cols + pad
#define PSTRIDE2 144   // 128 cols + pad

// ---------------------------------------------------------------------------
// Utility kernels
// ---------------------------------------------------------------------------

__global__ void zero_f32_kernel(float* __restrict__ p, long long n) {
  long long i = (long long)blockIdx.x * blockDim.x + threadIdx.x;
  long long stride = (long long)gridDim.x * blockDim.x;
  for (; i < n; i += stride) p[i] = 0.0f;
}

__global__ void edge_deg_kernel(const int* __restrict__ dst,
                                float* __restrict__ deg, int nE) {
  int i = blockIdx.x * blockDim.x + threadIdx.x;
  int stride = gridDim.x * blockDim.x;
  for (; i < nE; i += stride) atomicAdd(&deg[dst[i]], 1.0f);
}

__global__ void batch_cnt_kernel(const int* __restrict__ batch,
                                 float* __restrict__ cnt, int nN) {
  int i = blockIdx.x * blockDim.x + threadIdx.x;
  int stride = gridDim.x * blockDim.x;
  for (; i < nN; i += stride) atomicAdd(&cnt[batch[i]], 1.0f);
}

// One wave per edge: lanes cover 64 channels in two coalesced 128B strips.
// Feature tables are L2-resident (25.6 MB << 192 MB L2), so the gather and
// the atomic scatter-add both stay on-chip.
__global__ void __launch_bounds__(256)
scatter_add_kernel(const float* __restrict__ feat, const int* __restrict__ src,
                   const int* __restrict__ dst, float* __restrict__ agg, int nE) {
  const int lane = threadIdx.x & 31;
  int wv = blockIdx.x * (blockDim.x >> 5) + (threadIdx.x >> 5);
  const int nW = gridDim.x * (blockDim.x >> 5);
  for (int e = wv; e < nE; e += nW) {
    const int s = src[e];
    const int d = dst[e];
    const float* fr = feat + (size_t)s * 64;
    float* ar = agg + (size_t)d * 64;
    float v0 = fr[lane];
    float v1 = fr[lane + 32];
    atomicAdd(ar + lane, v0);
    atomicAdd(ar + lane + 32, v1);
  }
}

// ---------------------------------------------------------------------------
// conv1: h = relu( (agg/deg) @ Wl1^T + b1 + x @ Wr1^T ), 64 -> 64 channels.
// One wave per 16-node tile; 4 f32 accumulators (16x16) cover 64 outputs.
// A fragments preloaded per phase (batched global issue, single wait);
// B fragments are single 8B conflict-free LDS loads.
// ---------------------------------------------------------------------------
__global__ void __launch_bounds__(256)
conv1_wmma_kernel(const float* __restrict__ x, const float* __restrict__ agg,
                  const float* __restrict__ deg,
                  const float* __restrict__ Wl, const float* __restrict__ Wr,
                  const float* __restrict__ bias, float* __restrict__ h) {
  __shared__ float sWl[32 * PSTRIDE1 * 2];
  __shared__ float sWr[32 * PSTRIDE1 * 2];
  for (int idx = threadIdx.x; idx < 32 * 64; idx += blockDim.x) {
    const int row = idx >> 6;   // k>>1
    const int n = idx & 63;     // output column
    const int d = (row * PSTRIDE1 + n) * 2;
    const int s = n * 64 + row * 2;  // W row-major [out=n][in=k]
    sWl[d] = Wl[s];
    sWl[d + 1] = Wl[s + 1];
    sWr[d] = Wr[s];
    sWr[d + 1] = Wr[s + 1];
  }
  __syncthreads();

  const int lane = threadIdx.x & 31;
  const int wave = threadIdx.x >> 5;
  const int tile = blockIdx.x * 8 + wave;
  const int m0 = tile * 16;
  if (m0 >= N_NODES) return;  // wave-uniform

  const int lrow = lane & 15;
  const int kh = lane >> 4;       // K-half select (ISA A/B fragment layout)
  const int khalf = kh << 1;      // 0 or 2
  const int m = m0 + lrow;
  const int mclamp = (m < N_NODES) ? m : (N_NODES - 1);
  const float valid = (m < N_NODES) ? 1.0f : 0.0f;
  const float invd = valid / fmaxf(deg[mclamp], 1.0f);

  const v8f vz = {0, 0, 0, 0, 0, 0, 0, 0};
  v8f acc[4] = {vz, vz, vz, vz};

  const float* aggRow = agg + (size_t)mclamp * 64;
  const float* xRow = x + (size_t)mclamp * 64;

  v2f afrag[16];

  // ---- Phase 1: (agg/deg) @ Wl^T ----
#pragma unroll
  for (int kc = 0; kc < 16; ++kc) {
    v2f a = *(const v2f*)(aggRow + kc * 4 + khalf);
    afrag[kc] = a * invd;
  }
#pragma unroll
  for (int kc = 0; kc < 16; ++kc) {
    const float* bp = &sWl[((2 * kc + kh) * PSTRIDE1) * 2];
#pragma unroll
    for (int nb = 0; nb < 4; ++nb) {
      const v2f b = *(const v2f*)(bp + (nb * 16 + lrow) * 2);
      acc[nb] = wmma_f32_16x16x4(afrag[kc], b, acc[nb]);
    }
  }

  // ---- Phase 2: x @ Wr^T (accumulate into same C) ----
#pragma unroll
  for (int kc = 0; kc < 16; ++kc) {
    v2f a = *(const v2f*)(xRow + kc * 4 + khalf);
    afrag[kc] = a * valid;
  }
#pragma unroll
  for (int kc = 0; kc < 16; ++kc) {
    const float* bp = &sWr[((2 * kc + kh) * PSTRIDE1) * 2];
#pragma unroll
    for (int nb = 0; nb < 4; ++nb) {
      const v2f b = *(const v2f*)(bp + (nb * 16 + lrow) * 2);
      acc[nb] = wmma_f32_16x16x4(afrag[kc], b, acc[nb]);
    }
  }

  // Epilogue: +bias, relu, store. C layout: vgpr i, lanes0-15 -> M=i, N=lane;
  // lanes16-31 -> M=8+i, N=lane-16.
  const int mrow0 = kh * 8;
#pragma unroll
  for (int nb = 0; nb < 4; ++nb) {
    const int col = nb * 16 + lrow;
    const float bv = bias[col];
#pragma unroll
    for (int i = 0; i < 8; ++i) {
      const int mm = m0 + mrow0 + i;
      if (mm < N_NODES) {
        h[(size_t)mm * 64 + col] = fmaxf(acc[nb][i] + bv, 0.0f);
      }
    }
  }
}

// ---------------------------------------------------------------------------
// conv2: (agg2/deg) @ Wl2^T + b2 + h @ Wr2^T, 64 -> 128 channels, fused with
// global mean pool (run-length combined atomics; batch is sorted).
// 8 accumulators per wave. Wl2 pairs in LDS; Wr2 pairs from global (L2-hot).
// ---------------------------------------------------------------------------
__global__ void __launch_bounds__(256)
conv2_wmma_pool_kernel(const float* __restrict__ hfeat,
                       const float* __restrict__ agg,
                       const float* __restrict__ deg,
                       const float* __restrict__ Wl, const float* __restrict__ Wr,
                       const float* __restrict__ bias,
                       const int* __restrict__ batch,
                       float* __restrict__ pooled) {
  __shared__ float sWl[32 * PSTRIDE2 * 2];  // 36 KB
  for (int idx = threadIdx.x; idx < 32 * 128; idx += blockDim.x) {
    const int row = idx >> 7;    // k>>1
    const int n = idx & 127;     // output column
    const int d = (row * PSTRIDE2 + n) * 2;
    const int s = n * 64 + row * 2;  // Wl2 row-major [128][64]
    sWl[d] = Wl[s];
    sWl[d + 1] = Wl[s + 1];
  }
  __syncthreads();

  const int lane = threadIdx.x & 31;
  const int wave = threadIdx.x >> 5;
  const int tile = blockIdx.x * 8 + wave;
  const int m0 = tile * 16;
  if (m0 >= N_NODES) return;  // wave-uniform

  const int lrow = lane & 15;
  const int kh = lane >> 4;
  const int khalf = kh << 1;
  const int m = m0 + lrow;
  const int mclamp = (m < N_NODES) ? m : (N_NODES - 1);
  const float valid = (m < N_NODES) ? 1.0f : 0.0f;
  const float invd = valid / fmaxf(deg[mclamp], 1.0f);

  const v8f vz = {0, 0, 0, 0, 0, 0, 0, 0};
  v8f acc[8] = {vz, vz, vz, vz, vz, vz, vz, vz};

  const float* aggRow = agg + (size_t)mclamp * 64;
  const float* hRow = hfeat + (size_t)mclamp * 64;

  v2f afrag[16];

  // ---- Phase 1: (agg2/deg) @ Wl2^T ----
#pragma unroll
  for (int kc = 0; kc < 16; ++kc) {
    v2f a = *(const v2f*)(aggRow + kc * 4 + khalf);
    afrag[kc] = a * invd;
  }
#pragma unroll
  for (int kc = 0; kc < 16; ++kc) {
    const float* bp = &sWl[((2 * kc + kh) * PSTRIDE2) * 2];
#pragma unroll
    for (int nb = 0; nb < 8; ++nb) {
      const v2f b = *(const v2f*)(bp + (nb * 16 + lrow) * 2);
      acc[nb] = wmma_f32_16x16x4(afrag[kc], b, acc[nb]);
    }
  }

  // ---- Phase 2: h @ Wr2^T (Wr2 row-major [128][64]; pair loads, L2-hot) ----
#pragma unroll
  for (int kc = 0; kc < 16; ++kc) {
    v2f a = *(const v2f*)(hRow + kc * 4 + khalf);
    afrag[kc] = a * valid;
  }
#pragma unroll
  for (int kc = 0; kc < 16; ++kc) {
    const int k = kc * 4 + khalf;
#pragma unroll
    for (int nb = 0; nb < 8; ++nb) {
      const int col = nb * 16 + lrow;
      const v2f b = *(const v2f*)(Wr + col * 64 + k);
      acc[nb] = wmma_f32_16x16x4(afrag[kc], b, acc[nb]);
    }
  }

  // Epilogue: +bias, fused graph-mean-pool. batch is sorted, so run-length
  // combine equal-graph partial sums before hitting L2 atomics (~8x fewer).
  const int mrow0 = kh * 8;
  int gs[8];
#pragma unroll
  for (int i = 0; i < 8; ++i) {
    const int mm = m0 + mrow0 + i;
    gs[i] = (mm < N_NODES) ? batch[mm] : -1;
  }
#pragma unroll
  for (int nb = 0; nb < 8; ++nb) {
    const int col = nb * 16 + lrow;
    const float bv = bias[col];
    int curg = -1;
    float sum = 0.0f;
#pragma unroll
    for (int i = 0; i < 8; ++i) {
      if (gs[i] >= 0) {
        const float v = acc[nb][i] + bv;
        if (gs[i] != curg) {
          if (curg >= 0) atomicAdd(&pooled[(size_t)curg * OUT_CH + col], sum);
          curg = gs[i];
          sum = v;
        } else {
          sum += v;
        }
      }
    }
    if (curg >= 0) atomicAdd(&pooled[(size_t)curg * OUT_CH + col], sum);
  }
}

__global__ void finalize_kernel(const float* __restrict__ pooled,
                                const float* __restrict__ cnt,
                                float* __restrict__ out) {
  int i = blockIdx.x * blockDim.x + threadIdx.x;
  if (i < N_GRAPHS * OUT_CH) {
    out[i] = pooled[i] / fmaxf(cnt[i >> 7], 1.0f);
  }
}

// ---------------------------------------------------------------------------
// Launcher (graph-capture safe: stream-ordered kernels only)
// ---------------------------------------------------------------------------
extern "C" void kernel_launch(void* const* d_in, const int* in_sizes, int n_in,
                              void* d_out, int out_size, void* d_ws, size_t ws_size,
                              hipStream_t stream) {
  const float* x   = (const float*)d_in[0];
  const int*   ei  = (const int*)d_in[1];  // [2, E] flat
  const int*   bat = (const int*)d_in[2];
  const float* Wl1 = (const float*)d_in[3];
  const float* Wr1 = (const float*)d_in[4];
  const float* b1  = (const float*)d_in[5];
  const float* Wl2 = (const float*)d_in[6];
  const float* Wr2 = (const float*)d_in[7];
  const float* b2  = (const float*)d_in[8];
  float* out = (float*)d_out;

  const int* src = ei;
  const int* dst = ei + N_EDGES;

  // Workspace layout (floats)
  float* ws = (float*)d_ws;
  const long long DEG_PAD = 100352;  // N_NODES rounded up to 1024
  float* deg    = ws;
  float* agg    = deg + DEG_PAD;                  // [N, 64]
  float* h      = agg + (long long)N_NODES * 64;  // [N, 64]
  float* pooled = h + (long long)N_NODES * 64;    // [128, 128]
  float* cnt    = pooled + (long long)N_GRAPHS * OUT_CH;
  const long long ws_total = DEG_PAD + 2LL * N_NODES * 64 +
                             (long long)N_GRAPHS * OUT_CH + N_GRAPHS;

  const int nodeTiles = (N_NODES + 15) / 16;
  const int convBlocks = (nodeTiles + 7) / 8;  // 8 waves / block

  // 1) zero all workspace (deg, agg, h, pooled, cnt)
  zero_f32_kernel<<<4096, 256, 0, stream>>>(ws, ws_total);
  // 2) degrees + per-graph counts
  edge_deg_kernel<<<2048, 256, 0, stream>>>(dst, deg, N_EDGES);
  batch_cnt_kernel<<<512, 256, 0, stream>>>(bat, cnt, N_NODES);
  // 3) agg1 = scatter_sum(x[src] -> dst)
  scatter_add_kernel<<<2048, 256, 0, stream>>>(x, src, dst, agg, N_EDGES);
  // 4) conv1 (WMMA) -> h
  conv1_wmma_kernel<<<convBlocks, 256, 0, stream>>>(x, agg, deg, Wl1, Wr1, b1, h);
  // 5) re-zero agg, then agg2 = scatter_sum(h[src] -> dst)
  zero_f32_kernel<<<4096, 256, 0, stream>>>(agg, (long long)N_NODES * 64);
  scatter_add_kernel<<<2048, 256, 0, stream>>>(h, src, dst, agg, N_EDGES);
  // 6) conv2 (WMMA) fused with graph-mean-pool accumulation
  conv2_wmma_pool_kernel<<<convBlocks, 256, 0, stream>>>(h, agg, deg, Wl2, Wr2,
                                                         b2, bat, pooled);
  // 7) out = pooled / max(cnt, 1)
  finalize_kernel<<<64, 256, 0, stream>>>(pooled, cnt, out);
}